// PointerSeq2Seq_53154515256337
// MI455X (gfx1250) — compile-verified
//
#include <hip/hip_runtime.h>
#include <hip/hip_bf16.h>
#include <math.h>

#define B_    8
#define T_    128
#define S_    128
#define V_    32000
#define P_    100
#define VEXT_ (V_ + P_)
#define U_    512
#define LU_   1024
#define BT_   (B_ * T_)

typedef __attribute__((ext_vector_type(2))) float v2f;
typedef __attribute__((ext_vector_type(8))) float v8f;

// ---------------------------------------------------------------------------
// Kernel 1: gate GEMV via V_WMMA_F32_16X16X4_F32.
// One wave (32 lanes) computes sigmoid gates for 16 rows.
// A tile (16x4 f32): lane l<16 -> row l, K={k,k+1}; lane l>=16 -> row l-16, K={k+2,k+3}
// B tile (4x16 f32): weight values broadcast across all N columns.
// D (16x16): every column holds the same dot products; we read column N=0
// from lane 0 (M=0..7 in VGPR 0..7) and lane 16 (M=8..15).
// ---------------------------------------------------------------------------
__device__ __forceinline__ v8f gate_accum(const float* __restrict__ X,
                                          const float* __restrict__ W,
                                          int len, int row, int khalf, v8f acc) {
  const float* rowp = X + (size_t)row * (size_t)len;
  #pragma unroll 4
  for (int k = 0; k < len; k += 4) {
    int kk = k + khalf;
    v2f a, b;
    a.x = rowp[kk];
    a.y = rowp[kk + 1];
    b.x = W[kk];
    b.y = W[kk + 1];
    acc = __builtin_amdgcn_wmma_f32_16x16x4_f32(false, a, false, b,
                                                (short)0, acc, false, false);
  }
  return acc;
}

__global__ void gate_kernel(const float* __restrict__ contexts,
                            const float* __restrict__ states,
                            const float* __restrict__ inputs,
                            const float* __restrict__ Wc,
                            const float* __restrict__ Ws,
                            const float* __restrict__ Wi,
                            const float* __restrict__ bi,
                            float* __restrict__ gate) {
  const int lane    = threadIdx.x;          // 0..31, full wave active (EXEC all 1s)
  const int rowBase = blockIdx.x * 16;
  const int m       = lane & 15;
  const int khalf   = (lane >> 4) << 1;     // 0 or 2
  const int row     = rowBase + m;

  v8f acc = {};
  acc = gate_accum(contexts, Wc, U_,  row, khalf, acc);
  acc = gate_accum(states,   Ws, LU_, row, khalf, acc);
  acc = gate_accum(inputs,   Wi, U_,  row, khalf, acc);

  const float bias = bi[0];
  if ((lane & 15) == 0) {                   // lanes 0 and 16 hold column N=0
    const int mb = (lane >> 4) * 8;
    #pragma unroll
    for (int j = 0; j < 8; ++j) {
      float z = acc[j] + bias;
      gate[rowBase + mb + j] = 1.0f / (1.0f + __expf(-z));
    }
  }
}

// ---------------------------------------------------------------------------
// Kernel 2: fused softmax * gate, streamed with float4 (GLOBAL_LOAD/STORE_B128).
// One 256-thread block per row; online max/sum pass (HBM read), scaled-write
// pass (logits re-read hits L2: 131 MB tensor < 192 MB L2). Also zero-fills
// the 100-wide pad region so the scatter kernel can accumulate into it.
// ---------------------------------------------------------------------------
__global__ void __launch_bounds__(256)
softmax_gate_kernel(const float* __restrict__ logits,
                    const float* __restrict__ gate,
                    float* __restrict__ out) {
  const int tid = threadIdx.x;
  const int row = blockIdx.x;
  const float4* x4 = (const float4*)(logits + (size_t)row * V_);

  float m = -INFINITY, s = 0.0f;
  for (int i = tid; i < V_ / 4; i += 256) {
    float4 v = x4[i];
    float lm = fmaxf(fmaxf(v.x, v.y), fmaxf(v.z, v.w));
    if (lm > m) { s *= __expf(m - lm); m = lm; }
    s += __expf(v.x - m) + __expf(v.y - m) + __expf(v.z - m) + __expf(v.w - m);
  }

  __shared__ float sm[256];
  __shared__ float ss[256];
  sm[tid] = m; ss[tid] = s;
  __syncthreads();
  for (int off = 128; off > 0; off >>= 1) {
    if (tid < off) {
      float m2 = sm[tid + off], s2 = ss[tid + off];
      float M  = fmaxf(sm[tid], m2);
      ss[tid]  = ss[tid] * __expf(sm[tid] - M) + s2 * __expf(m2 - M);
      sm[tid]  = M;
    }
    __syncthreads();
  }
  const float M     = sm[0];
  const float scale = gate[row] / ss[0];    // softmax * switch fused

  float4* o4 = (float4*)(out + (size_t)row * VEXT_);   // 32100*4 bytes is 16B-aligned
  for (int i = tid; i < V_ / 4; i += 256) {
    float4 v = x4[i];
    float4 r;
    r.x = __expf(v.x - M) * scale;
    r.y = __expf(v.y - M) * scale;
    r.z = __expf(v.z - M) * scale;
    r.w = __expf(v.w - M) * scale;
    o4[i] = r;
  }
  if (tid < P_) out[(size_t)row * VEXT_ + V_ + tid] = 0.0f;
}

// ---------------------------------------------------------------------------
// Kernel 3: scatter-add of attention mass (128 entries/row, duplicate indices
// possible -> GLOBAL_ATOMIC_ADD_F32; out rows are L2-resident). Separate launch
// guarantees ordering vs. the base writes of kernel 2.
// ---------------------------------------------------------------------------
__global__ void __launch_bounds__(256)
scatter_kernel(const float* __restrict__ attn,
               const int* __restrict__ texts,
               const float* __restrict__ gate,
               float* __restrict__ out) {
  int idx = blockIdx.x * blockDim.x + threadIdx.x;   // [0, BT*S)
  if (idx >= BT_ * S_) return;
  int row = idx >> 7;                                // S_ == 128
  int s   = idx & (S_ - 1);
  int b   = row >> 7;                                // T_ == 128
  float w = attn[idx] * (1.0f - gate[row]);
  int v   = texts[b * S_ + s];
  atomicAdd(out + (size_t)row * VEXT_ + v, w);
}

// ---------------------------------------------------------------------------
extern "C" void kernel_launch(void* const* d_in, const int* in_sizes, int n_in,
                              void* d_out, int out_size, void* d_ws, size_t ws_size,
                              hipStream_t stream) {
  const float* logits   = (const float*)d_in[0];
  const float* attn     = (const float*)d_in[1];
  const float* contexts = (const float*)d_in[2];
  const float* states   = (const float*)d_in[3];
  const float* inputs   = (const float*)d_in[4];
  const float* Wc       = (const float*)d_in[5];
  const float* Ws       = (const float*)d_in[6];
  const float* Wi       = (const float*)d_in[7];
  const float* bi       = (const float*)d_in[8];
  const int*   texts    = (const int*)d_in[9];
  float* out  = (float*)d_out;
  float* gate = (float*)d_ws;                        // 1024 floats of scratch

  gate_kernel<<<BT_ / 16, 32, 0, stream>>>(contexts, states, inputs,
                                           Wc, Ws, Wi, bi, gate);
  softmax_gate_kernel<<<BT_, 256, 0, stream>>>(logits, gate, out);
  scatter_kernel<<<(BT_ * S_) / 256, 256, 0, stream>>>(attn, texts, gate, out);
}